// ICS_73366631350293
// MI455X (gfx1250) — compile-verified
//
#include <hip/hip_runtime.h>
#include <hip/hip_bf16.h>

// CDNA5 / gfx1250 implementation of the XCA "transposed attention" block.
// GEMM-shaped work (qkv 1x1 conv, Q^T*K, V*A, proj 1x1 conv) runs on
// V_WMMA_F32_16X16X4_F32 (fp32 in / fp32 out, wave32). Shared operands are
// staged in LDS with the gfx1250 async global->LDS path
// (global_load_async_to_lds_b128 + s_wait_asynccnt), double-buffered in the
// conv GEMM. LDS rows are padded to 132 words so strided reads hit distinct
// banks.
//
// ws layout (floats):
//   qkv1 = ws                   : 8*384*16384  (conv1x1 out; reused as attention out)
//   qkv2 = ws + 8*384*16384     : 8*384*16384  (dwconv out = q|k|v, normalized in place)
// Required ws_size >= 2 * 8*384*16384 * 4 bytes = 384 MB.

typedef __attribute__((ext_vector_type(2))) float v2f;
typedef __attribute__((ext_vector_type(8))) float v8f;

#define HWSZ 16384   // 128*128 pixels per plane
#define WD   128
#define BATCH 8
#define LPAD 132     // padded LDS row stride (16B-aligned, bank-conflict free)

static __device__ __forceinline__ v8f wmma_f32(v2f a, v2f b, v8f c) {
    // D(16x16) = A(16x4) * B(4x16) + C  -- fp32 WMMA, 8-arg form
    return __builtin_amdgcn_wmma_f32_16x16x4_f32(
        /*neg_a=*/false, a, /*neg_b=*/false, b,
        /*c_mod=*/(short)0, c, /*reuse_a=*/false, /*reuse_b=*/false);
}

static __device__ __forceinline__ v8f vzero8() {
    v8f z = {0.f, 0.f, 0.f, 0.f, 0.f, 0.f, 0.f, 0.f};
    return z;
}

// Async DMA of 16 bytes global -> LDS (per lane). Tracked by ASYNCcnt.
static __device__ __forceinline__ void async_copy_b128(void* lds_dst, const void* gsrc) {
    asm volatile("global_load_async_to_lds_b128 %0, %1, off"
                 :: "v"((unsigned)(uintptr_t)lds_dst), "v"(gsrc)
                 : "memory");
}
static __device__ __forceinline__ void wait_async_le(int n) {
    if (n == 0)  asm volatile("s_wait_asynccnt 0x0"  ::: "memory");
    else         asm volatile("s_wait_asynccnt 0x10" ::: "memory"); // <=16
}

// ---------------------------------------------------------------------------
// 1x1 conv as GEMM: Y[b][o][p] = sum_c W[o][c] * X[b][c][p] + bias[o]
// grid = (P/128, Cout/128, B), block = 256 (8 waves).
// X tiles (shared by all 8 waves) are async-staged into LDS, double-buffered
// in 32-row K-chunks. Weights stream from global (tiny, L2-resident).
// ---------------------------------------------------------------------------
__global__ __launch_bounds__(256) void conv1x1_wmma_kernel(
    const float* __restrict__ X, const float* __restrict__ W,
    const float* __restrict__ bias, float* __restrict__ Y,
    int Cin, int Cout)
{
    __shared__ float Xs[2][32 * LPAD];   // 2 x ~16.5 KB, double buffered

    const int p0    = blockIdx.x * 128;
    const int o_blk = blockIdx.y * 128;
    const int b     = blockIdx.z;
    const int wave  = threadIdx.x >> 5;
    const int lane  = threadIdx.x & 31;
    const int m     = lane & 15;             // M (or N) index within tile
    const int koff  = (lane >> 4) << 1;      // lane half selects K {0,1} vs {2,3}
    const int hi8   = (lane >> 4) << 3;      // C/D row offset for upper lane half
    const int o_base = o_blk + wave * 16;

    v8f acc[8];
#pragma unroll
    for (int t = 0; t < 8; ++t) acc[t] = vzero8();

    // stage one 32(c) x 128(p) chunk of X into Xs[buf]
    auto stage_chunk = [&](int buf, int cbase) {
#pragma unroll
        for (int i = 0; i < 4; ++i) {
            const int idx = (int)threadIdx.x + i * 256;   // 0..1023 float4s
            const int cl  = idx >> 5;                     // 0..31
            const int q4  = (idx & 31) << 2;              // 0..124
            async_copy_b128(&Xs[buf][cl * LPAD + q4],
                            X + ((size_t)b * Cin + cbase + cl) * HWSZ + p0 + q4);
        }
    };

    stage_chunk(0, 0);
    int buf = 0;
    const float* wrow = W + (size_t)(o_base + m) * Cin;

    for (int cc = 0; cc < Cin; cc += 32) {
        wait_async_le(0);
        __syncthreads();                       // chunk `buf` visible to all waves
        if (cc + 32 < Cin) stage_chunk(buf ^ 1, cc + 32);

        const float* Xsb = &Xs[buf][0];
        for (int c = 0; c < 32; c += 4) {
            v2f a;
            a.x = wrow[cc + c + koff];
            a.y = wrow[cc + c + koff + 1];
#pragma unroll
            for (int t = 0; t < 8; ++t) {
                v2f bb;
                bb.x = Xsb[(c + koff) * LPAD + t * 16 + m];
                bb.y = Xsb[(c + koff + 1) * LPAD + t * 16 + m];
                acc[t] = wmma_f32(a, bb, acc[t]);
            }
        }
        buf ^= 1;
    }

#pragma unroll
    for (int t = 0; t < 8; ++t) {
#pragma unroll
        for (int r = 0; r < 8; ++r) {
            const int o = o_base + r + hi8;
            Y[((size_t)b * Cout + o) * HWSZ + p0 + t * 16 + m] = acc[t][r] + bias[o];
        }
    }
}

// ---------------------------------------------------------------------------
// Depthwise 3x3 conv, padding 1, + bias. One thread per output element.
// ---------------------------------------------------------------------------
__global__ __launch_bounds__(256) void dwconv3x3_kernel(
    const float* __restrict__ X, const float* __restrict__ W9,
    const float* __restrict__ bias, float* __restrict__ Y)
{
    const size_t idx = (size_t)blockIdx.x * 256 + threadIdx.x;  // over B*384*HW
    const int wx = (int)(idx & 127);
    const int hy = (int)((idx >> 7) & 127);
    const int ch = (int)((idx >> 14) % 384);
    const float* xp = X + (idx & ~(size_t)(HWSZ - 1));          // plane base
    const float* wp = W9 + ch * 9;

    float s = bias[ch];
#pragma unroll
    for (int dy = -1; dy <= 1; ++dy) {
        const int yy = hy + dy;
        if (yy < 0 || yy > 127) continue;
#pragma unroll
        for (int dx = -1; dx <= 1; ++dx) {
            const int xx = wx + dx;
            if (xx < 0 || xx > 127) continue;
            s += wp[(dy + 1) * 3 + (dx + 1)] * xp[yy * WD + xx];
        }
    }
    Y[idx] = s;
}

// ---------------------------------------------------------------------------
// L2-normalize q and k along the H axis, in place.
// One block per (b, ch') with ch' in [0,256): q channels 0..127, k 128..255.
// ---------------------------------------------------------------------------
__global__ __launch_bounds__(128) void l2norm_kernel(float* __restrict__ QK)
{
    const int blk = blockIdx.x;
    const int b   = blk >> 8;
    const int ch  = blk & 255;
    float* base = QK + ((size_t)b * 384 + ch) * HWSZ;
    const int t = threadIdx.x;

    float ss = 0.f;
    for (int h = 0; h < 128; ++h) {
        const float v = base[h * WD + t];
        ss += v * v;
    }
    const float scale = 1.0f / fmaxf(sqrtf(ss), 1e-12f);
    for (int h = 0; h < 128; ++h)
        base[h * WD + t] *= scale;
}

// ---------------------------------------------------------------------------
// Per-slice transposed attention. One block per (b, ch), ch in [0,128).
// K and V slices are async-DMA'd into padded LDS up front (K needed first,
// V overlaps phase 1 + softmax). S lives in LDS between the two WMMA passes.
// Total LDS: 3 * 128 * 132 * 4B ~= 198 KB of the 320 KB WGP budget.
// ---------------------------------------------------------------------------
__global__ __launch_bounds__(256) void xattn_kernel(
    const float* __restrict__ QKV, const float* __restrict__ temp4,
    float* __restrict__ O)
{
    __shared__ float S [128 * LPAD];
    __shared__ float Ks[128 * LPAD];
    __shared__ float Vs[128 * LPAD];

    const int b    = blockIdx.x >> 7;
    const int ch   = blockIdx.x & 127;
    const float tau = temp4[ch >> 5];                 // head = ch/32
    const float* Q = QKV + ((size_t)b * 384 + ch) * HWSZ;
    const float* K = Q + (size_t)128 * HWSZ;
    const float* V = Q + (size_t)256 * HWSZ;

    const int wave = threadIdx.x >> 5;
    const int lane = threadIdx.x & 31;
    const int m    = lane & 15;
    const int koff = (lane >> 4) << 1;
    const int hi8  = (lane >> 4) << 3;

    // ---- async-stage K then V into LDS (16 + 16 DMA ops per wave) ----
#pragma unroll
    for (int i = 0; i < 16; ++i) {
        const int idx = (int)threadIdx.x + i * 256;   // 0..4095 float4s
        const int row = idx >> 5;
        const int q4  = (idx & 31) << 2;
        async_copy_b128(&Ks[row * LPAD + q4], K + row * WD + q4);
    }
#pragma unroll
    for (int i = 0; i < 16; ++i) {
        const int idx = (int)threadIdx.x + i * 256;
        const int row = idx >> 5;
        const int q4  = (idx & 31) << 2;
        async_copy_b128(&Vs[row * LPAD + q4], V + row * WD + q4);
    }
    wait_async_le(16);      // K copies (first 16, in-order) complete; V in flight
    __syncthreads();

    // ---- Phase 1: S[j][k] = tau * sum_i Q[i][j] * K[i][k] ----
    {
        const int j0 = wave * 16;
        v8f acc[8];
#pragma unroll
        for (int kt = 0; kt < 8; ++kt) acc[kt] = vzero8();

        for (int i = 0; i < 128; i += 4) {
            v2f a;                       // A[m=j][kk=i] = Q[i][j0+m]
            a.x = Q[(i + koff) * WD + j0 + m];
            a.y = Q[(i + koff + 1) * WD + j0 + m];
#pragma unroll
            for (int kt = 0; kt < 8; ++kt) {
                v2f bb;                  // B[kk=i][n=k] = K[i][kt*16+m]
                bb.x = Ks[(i + koff) * LPAD + kt * 16 + m];
                bb.y = Ks[(i + koff + 1) * LPAD + kt * 16 + m];
                acc[kt] = wmma_f32(a, bb, acc[kt]);
            }
        }
#pragma unroll
        for (int kt = 0; kt < 8; ++kt)
#pragma unroll
            for (int r = 0; r < 8; ++r)
                S[(j0 + r + hi8) * LPAD + kt * 16 + m] = acc[kt][r] * tau;
    }
    __syncthreads();

    // ---- Phase 2: softmax over j for each column k ----
    if (threadIdx.x < 128) {
        const int k = threadIdx.x;
        float mx = -3.402823466e+38f;
        for (int j = 0; j < 128; ++j) mx = fmaxf(mx, S[j * LPAD + k]);
        float sum = 0.f;
        for (int j = 0; j < 128; ++j) {
            const float e = __expf(S[j * LPAD + k] - mx);
            S[j * LPAD + k] = e;
            sum += e;
        }
        const float inv = 1.0f / sum;
        for (int j = 0; j < 128; ++j) S[j * LPAD + k] *= inv;
    }
    wait_async_le(0);       // V copies complete
    __syncthreads();

    // ---- Phase 3: Out[i][k] = sum_j V[i][j] * S[j][k] ----
    {
        const int i0 = wave * 16;
        v8f acc[8];
#pragma unroll
        for (int kt = 0; kt < 8; ++kt) acc[kt] = vzero8();

        for (int j = 0; j < 128; j += 4) {
            v2f a;                       // A[m=i][kk=j] = V[i0+m][j]  (from LDS)
            a.x = Vs[(i0 + m) * LPAD + j + koff];
            a.y = Vs[(i0 + m) * LPAD + j + koff + 1];
#pragma unroll
            for (int kt = 0; kt < 8; ++kt) {
                v2f bb;                  // B[kk=j][n=k] = S[j][kt*16+m]
                bb.x = S[(j + koff) * LPAD + kt * 16 + m];
                bb.y = S[(j + koff + 1) * LPAD + kt * 16 + m];
                acc[kt] = wmma_f32(a, bb, acc[kt]);
            }
        }
        float* Ob = O + ((size_t)b * 128 + ch) * HWSZ;
#pragma unroll
        for (int kt = 0; kt < 8; ++kt)
#pragma unroll
            for (int r = 0; r < 8; ++r)
                Ob[(i0 + r + hi8) * WD + kt * 16 + m] = acc[kt][r];
    }
}

// ---------------------------------------------------------------------------
extern "C" void kernel_launch(void* const* d_in, const int* in_sizes, int n_in,
                              void* d_out, int out_size, void* d_ws, size_t ws_size,
                              hipStream_t stream)
{
    (void)in_sizes; (void)n_in; (void)out_size; (void)ws_size;

    const float* x      = (const float*)d_in[0];
    const float* qkv_w  = (const float*)d_in[1];
    const float* qkv_b  = (const float*)d_in[2];
    const float* dw_w   = (const float*)d_in[3];
    const float* dw_b   = (const float*)d_in[4];
    const float* proj_w = (const float*)d_in[5];
    const float* proj_b = (const float*)d_in[6];
    const float* temp   = (const float*)d_in[7];
    float* out = (float*)d_out;

    float* ws   = (float*)d_ws;
    const size_t plane = (size_t)BATCH * 384 * HWSZ;   // 50,331,648 floats
    float* qkv1 = ws;                 // conv1x1 output
    float* qkv2 = ws + plane;         // dwconv output (q|k|v), normalized in place
    float* attn_out = qkv1;           // qkv1 is dead after dwconv -> reuse

    // qkv = conv1x1(x):  M=384, K=128, N=16384 per batch
    conv1x1_wmma_kernel<<<dim3(128, 3, BATCH), 256, 0, stream>>>(
        x, qkv_w, qkv_b, qkv1, 128, 384);

    // depthwise 3x3 + bias
    dwconv3x3_kernel<<<(int)((size_t)BATCH * 384 * HWSZ / 256), 256, 0, stream>>>(
        qkv1, dw_w, dw_b, qkv2);

    // L2-normalize q and k along H, in place
    l2norm_kernel<<<BATCH * 256, 128, 0, stream>>>(qkv2);

    // per-slice attention: 8*128 = 1024 blocks
    xattn_kernel<<<BATCH * 128, 256, 0, stream>>>(qkv2, temp, attn_out);

    // proj = conv1x1(out): M=128, K=128, N=16384 per batch
    conv1x1_wmma_kernel<<<dim3(128, 1, BATCH), 256, 0, stream>>>(
        attn_out, proj_w, proj_b, out, 128, 128);
}